// GCN_01_9689446220545
// MI455X (gfx1250) — compile-verified
//
#include <hip/hip_runtime.h>
#include <hip/hip_bf16.h>
#include <cstdint>

// ---------------------------------------------------------------------------
// GCN double-branch + h1^T h2 on MI455X (gfx1250, wave32, WMMA).
//   N=50000 nodes, E=800000 edges, F=256 features, all fp32.
// Aggregations: edge-parallel scatter with f32 global atomics (L2-resident,
//   h matrix = 51.2MB < 192MB L2). GEMMs: V_WMMA_F32_16X16X4_F32 (full fp32
//   fidelity; GEMM is not the bottleneck). Dense GEMM stages the W^T tile
//   in LDS (16KB/block) so all 8 waves share one B-operand fetch.
// ---------------------------------------------------------------------------

#define NF 256          // feature dim (fixed by the reference)
#define BN_EPS 1e-5f

typedef float v2f __attribute__((ext_vector_type(2)));
typedef float v8f __attribute__((ext_vector_type(8)));

__device__ __forceinline__ void atomAddF(float* p, float v) {
    // relaxed agent-scope fp32 add -> global_atomic_add_f32 on gfx1250
    __hip_atomic_fetch_add(p, v, __ATOMIC_RELAXED, __HIP_MEMORY_SCOPE_AGENT);
}

// ---------------- zero fill (float4 granularity) ----------------
__global__ void k_zero4(float4* __restrict__ p, long long n4) {
    long long i = (long long)blockIdx.x * blockDim.x + threadIdx.x;
    if (i < n4) { float4 z = {0.f, 0.f, 0.f, 0.f}; p[i] = z; }
}

// ---------------- edge-parallel scatter-add: out[dst[e]] += h[src[e]] -------
// 64 chunks of float4 per row; thread t -> (edge = t>>6, chunk = t&63).
// 32 consecutive lanes cover 512B contiguous within one row: fully coalesced.
__global__ void k_scatter_add(const float* __restrict__ h,
                              const int* __restrict__ src,
                              const int* __restrict__ dst,
                              float* __restrict__ out, long long total) {
    long long t = (long long)blockIdx.x * blockDim.x + threadIdx.x;
    if (t >= total) return;
    int e = (int)(t >> 6);
    int c = (int)(t & 63);
    int s = __ldg(&src[e]);
    int d = __ldg(&dst[e]);
    const float4 v = *(const float4*)(h + (size_t)s * NF + c * 4);
    float* o = out + (size_t)d * NF + c * 4;
    atomAddF(o + 0, v.x);
    atomAddF(o + 1, v.y);
    atomAddF(o + 2, v.z);
    atomAddF(o + 3, v.w);
}

// ---------------- out = A @ W^T + b  (A:[n,256], W:[256,256]) ---------------
// Block = (32,8): 8 waves, each owning one 16-row tile; blockIdx.y picks the
// 16-column tile. The W^T tile (B operand, 256x16 = 16KB) is staged in LDS
// once per block; stored as wtile[k][j] so B-fragment reads hit all 64 banks
// conflict-free ((16k + lane) word addressing).
// K marched in steps of 4 via V_WMMA_F32_16X16X4_F32.
__global__ void k_gemm_nt_bias(const float* __restrict__ A,
                               const float* __restrict__ W,
                               const float* __restrict__ bias,
                               float* __restrict__ out, int nrows) {
    __shared__ float wtile[NF][16];                  // B tile: wtile[k][j] = W[j0+j][k]
    const int lane = threadIdx.x;                    // 0..31
    const int wave = threadIdx.y;                    // 0..7
    const int tid  = wave * 32 + lane;               // 0..255
    const int j0 = blockIdx.y * 16;

    // ---- stage W^T tile: thread t loads row j = t>>4, 16-float k-chunk ----
    {
        const int j  = tid >> 4;
        const int kb = (tid & 15) * 16;
        const float* wr = W + (size_t)(j0 + j) * NF + kb;
#pragma unroll
        for (int i = 0; i < 16; i += 4) {
            float4 w4 = *(const float4*)(wr + i);    // 16B global load
            wtile[kb + i + 0][j] = w4.x;
            wtile[kb + i + 1][j] = w4.y;
            wtile[kb + i + 2][j] = w4.z;
            wtile[kb + i + 3][j] = w4.w;
        }
    }
    __syncthreads();

    const int m0 = (blockIdx.x * 8 + wave) * 16;
    if (m0 >= nrows) return;                         // wave-uniform tail exit
    const int half = lane >> 4;                      // K-pair selector (A/B layout)
    const int lr   = lane & 15;                      // M (for A) / N (for B)

    const float* arow = A + (size_t)(m0 + lr) * NF + 2 * half;

    v8f acc = {0.f, 0.f, 0.f, 0.f, 0.f, 0.f, 0.f, 0.f};
#pragma unroll 4
    for (int k = 0; k < NF; k += 4) {
        v2f a = *(const v2f*)(arow + k);             // A[m][k+2h .. +1], 8B aligned
        v2f b = { wtile[k + 2 * half][lr],           // B[k+2h][j], from LDS
                  wtile[k + 2 * half + 1][lr] };
        acc = __builtin_amdgcn_wmma_f32_16x16x4_f32(
            /*neg_a=*/false, a, /*neg_b=*/false, b,
            /*c_mod=*/(short)0, acc, /*reuse_a=*/false, /*reuse_b=*/false);
    }

    const float bj = __ldg(&bias[j0 + lr]);
#pragma unroll
    for (int r = 0; r < 8; ++r)            // C/D layout: row = r + 8*half, col = lr
        out[(size_t)(m0 + r + 8 * half) * NF + (j0 + lr)] = acc[r] + bj;
}

// ---------------- batchnorm: column sums / sum-of-squares -------------------
__global__ void k_bn_stats(const float* __restrict__ h,
                           float* __restrict__ sum, float* __restrict__ sumsq,
                           int n) {
    const int f = threadIdx.x;             // 256 threads = 256 features
    float s = 0.f, s2 = 0.f;
    for (int r = blockIdx.x; r < n; r += gridDim.x) {
        float v = h[(size_t)r * NF + f];   // coalesced across threads
        s += v; s2 += v * v;
    }
    atomAddF(&sum[f], s);
    atomAddF(&sumsq[f], s2);
}

// ---------------- batchnorm: normalize in place -----------------------------
__global__ void k_bn_apply(float* __restrict__ h,
                           const float* __restrict__ sum,
                           const float* __restrict__ sumsq,
                           const float* __restrict__ gamma,
                           const float* __restrict__ beta, int n) {
    long long i = (long long)blockIdx.x * blockDim.x + threadIdx.x;
    if (i >= (long long)n * NF) return;
    int f = (int)(i & (NF - 1));
    float invn = 1.0f / (float)n;
    float mean = sum[f] * invn;
    float var  = fmaxf(sumsq[f] * invn - mean * mean, 0.f);  // biased var
    float sc = gamma[f] * rsqrtf(var + BN_EPS);
    float sh = beta[f] - mean * sc;
    h[i] = h[i] * sc + sh;
}

// ---------------- out[256,256] += h1^T @ h2 (split-K WMMA) ------------------
// tile = blockIdx.x selects 16x16 output tile; each wave owns a K-chunk of
// KC nodes, accumulates with V_WMMA_F32_16X16X4_F32, then atomically folds
// its partial into d_out. All loads are 64B-contiguous per half-wave and
// L2-resident (each h buffer = 51.2MB).
#define KC 512
__global__ void k_gemm_tn_splitk(const float* __restrict__ h1,
                                 const float* __restrict__ h2,
                                 float* __restrict__ out, int n) {
    const int lane = threadIdx.x;
    const int wave = threadIdx.y;
    const int tile = blockIdx.x;                 // 0..255
    const int ti = (tile >> 4) * 16;             // output row tile (h1 feature)
    const int tj = (tile & 15) * 16;             // output col tile (h2 feature)
    int kBeg = (blockIdx.y * 8 + wave) * KC;
    if (kBeg >= n) return;                       // wave-uniform
    int kEnd = (kBeg + KC < n) ? (kBeg + KC) : n;  // chunk sizes stay %4==0 (n%4==0)
    const int half = lane >> 4;
    const int lr = lane & 15;

    v8f acc = {0.f, 0.f, 0.f, 0.f, 0.f, 0.f, 0.f, 0.f};
    for (int k = kBeg; k < kEnd; k += 4) {
        const int k0 = k + 2 * half;
        // A[m][kk] = h1[kk][ti+m]; B[kk][j] = h2[kk][tj+j]
        v2f a = { h1[(size_t)k0 * NF + ti + lr], h1[(size_t)(k0 + 1) * NF + ti + lr] };
        v2f b = { h2[(size_t)k0 * NF + tj + lr], h2[(size_t)(k0 + 1) * NF + tj + lr] };
        acc = __builtin_amdgcn_wmma_f32_16x16x4_f32(
            false, a, false, b, (short)0, acc, false, false);
    }
#pragma unroll
    for (int r = 0; r < 8; ++r)
        atomAddF(&out[(size_t)(ti + r + 8 * half) * NF + tj + lr], acc[r]);
}

// ---------------------------------------------------------------------------
// host-side orchestration
// ---------------------------------------------------------------------------
static inline void zero_f32(float* p, long long nfloats, hipStream_t s) {
    long long n4 = nfloats >> 2;
    int blk = 256;
    long long grid = (n4 + blk - 1) / blk;
    k_zero4<<<(unsigned)grid, blk, 0, s>>>((float4*)p, n4);
}

static void agg(const float* h, const int* src, const int* dst,
                float* out, int n, int E, hipStream_t s) {
    zero_f32(out, (long long)n * NF, s);
    long long total = (long long)E * 64;
    int blk = 256;
    long long grid = (total + blk - 1) / blk;
    k_scatter_add<<<(unsigned)grid, blk, 0, s>>>(h, src, dst, out, total);
}

static void gemm(const float* A, const float* W, const float* b,
                 float* out, int n, hipStream_t s) {
    int mtiles = (n + 15) / 16;
    dim3 grid((mtiles + 7) / 8, NF / 16);
    dim3 blk(32, 8);
    k_gemm_nt_bias<<<grid, blk, 0, s>>>(A, W, b, out, n);
}

static void run_branch(const float* feat, const int* src, const int* dst,
                       const float* W1, const float* b1,
                       const float* W2, const float* b2,
                       const float* W3, const float* b3,
                       const float* gamma, const float* beta,
                       float* H, float* T0, float* T1,
                       float* sums, float* sumsq,
                       int n, int E, hipStream_t s) {
    // layer 1: H = agg(agg(feat)) @ W1^T + b1
    agg(feat, src, dst, T0, n, E, s);
    agg(T0,   src, dst, T1, n, E, s);
    gemm(T1, W1, b1, H, n, s);
    // batchnorm (training mode, biased var) in place on H
    zero_f32(sums, 2 * NF, s);               // sums + sumsq contiguous
    k_bn_stats<<<512, NF, 0, s>>>(H, sums, sumsq, n);
    {
        long long tot = (long long)n * NF;
        int blk = 256;
        k_bn_apply<<<(unsigned)((tot + blk - 1) / blk), blk, 0, s>>>(
            H, sums, sumsq, gamma, beta, n);
    }
    // layer 2
    agg(H,  src, dst, T0, n, E, s);
    agg(T0, src, dst, T1, n, E, s);
    gemm(T1, W2, b2, H, n, s);
    // layer 3
    agg(H,  src, dst, T0, n, E, s);
    agg(T0, src, dst, T1, n, E, s);
    gemm(T1, W3, b3, H, n, s);
}

extern "C" void kernel_launch(void* const* d_in, const int* in_sizes, int n_in,
                              void* d_out, int out_size, void* d_ws, size_t ws_size,
                              hipStream_t stream) {
    const float* feature = (const float*)d_in[0];
    const int*   src1    = (const int*)d_in[1];
    const int*   dst1    = (const int*)d_in[2];
    const int*   src2    = (const int*)d_in[3];
    const int*   dst2    = (const int*)d_in[4];
    const float* W1 = (const float*)d_in[5];
    const float* b1 = (const float*)d_in[6];
    const float* W2 = (const float*)d_in[7];
    const float* b2 = (const float*)d_in[8];
    const float* W3 = (const float*)d_in[9];
    const float* b3 = (const float*)d_in[10];
    const float* gamma = (const float*)d_in[11];
    const float* beta  = (const float*)d_in[12];
    float* out = (float*)d_out;

    const int n = in_sizes[0] / NF;          // 50000
    const int E = in_sizes[1];               // 800000

    // workspace layout: 4 big [n,256] fp32 buffers + BN stats (2*256 floats)
    const size_t nf = (size_t)n * NF;
    float* ws = (float*)d_ws;
    float* H1 = ws;                // branch 1 result
    float* H2 = ws + nf;           // branch 2 result
    float* T0 = ws + 2 * nf;       // agg scratch
    float* T1 = ws + 3 * nf;       // agg scratch
    float* sums  = ws + 4 * nf;    // [256]
    float* sumsq = sums + NF;      // [256]
    (void)ws_size; (void)n_in; (void)out_size;

    run_branch(feature, src1, dst1, W1, b1, W2, b2, W3, b3, gamma, beta,
               H1, T0, T1, sums, sumsq, n, E, stream);
    run_branch(feature, src2, dst2, W1, b1, W2, b2, W3, b3, gamma, beta,
               H2, T0, T1, sums, sumsq, n, E, stream);

    // out = H1^T @ H2, split-K with fp32 atomics
    zero_f32(out, NF * NF, stream);
    {
        dim3 grid((NF / 16) * (NF / 16), (unsigned)((n + KC * 8 - 1) / (KC * 8)));
        dim3 blk(32, 8);
        k_gemm_tn_splitk<<<grid, blk, 0, stream>>>(H1, H2, out, n);
    }
}